// Encoder_51814485459365
// MI455X (gfx1250) — compile-verified
//
#include <hip/hip_runtime.h>

typedef __attribute__((ext_vector_type(2))) float        vf2;
typedef __attribute__((ext_vector_type(8))) float        vf8;
typedef __attribute__((ext_vector_type(4))) unsigned int v4u;
typedef __attribute__((ext_vector_type(8))) int          v8i;
typedef __attribute__((ext_vector_type(4))) int          v4i;

#define BB     64
#define MM     512
#define TT     8
#define EE     128
#define HOPSN  3
#define MSPLIT 8
#define MTILE  16

// Tensor Data Mover: arity differs across toolchains (see CDNA5_HIP.md).
#if defined(__clang_major__) && (__clang_major__ >= 23)
#define TDM_LOAD(g0, g1, g2, g3) \
    __builtin_amdgcn_tensor_load_to_lds((g0), (g1), (g2), (g3), (v8i){0,0,0,0,0,0,0,0}, 0)
#else
#define TDM_LOAD(g0, g1, g2, g3) \
    __builtin_amdgcn_tensor_load_to_lds((g0), (g1), (g2), (g3), 0)
#endif

static __device__ __forceinline__ vf8 wmma_f32_16x16x4(vf2 a, vf2 b, vf8 c) {
    // 8 args: (neg_a, A, neg_b, B, c_mod, C, reuse_a, reuse_b)
    return __builtin_amdgcn_wmma_f32_16x16x4_f32(false, a, false, b, (short)0, c,
                                                 false, false);
}

// TDM-gathered 16(m) x 128(e) tile of masked embedding-row sums into s_lds.
// Wave 0 issues one gather-mode TENSOR_LOAD_TO_LDS per m (8 rows x 512B each,
// 32-bit row indices in D# groups 2/3); all 128 threads then reduce the staged
// rows with the padding-idx mask. Caller must __syncthreads() after this
// before consuming s_lds, and before re-entering (staging reuse).
static __device__ __forceinline__ void gather_tile_tdm(
    const int* __restrict__ ctx_bm,        // [MTILE][TT] indices (block-uniform)
    const float* __restrict__ Ch,          // embedding table base [nwords][EE]
    int tid, unsigned int nwords,
    float (*rows)[TT][EE],                 // LDS staging [MTILE][TT][EE]
    float (*s_lds)[EE + 4]) {
    if (tid < 32) {                        // wave 0 issues the DMA descriptors
        const unsigned long long base = (unsigned long long)(uintptr_t)Ch;
        const unsigned int lds0 = (unsigned int)(uintptr_t)&rows[0][0][0];
        for (int m = 0; m < MTILE; ++m) {
            v4u g0;
            g0.x = 0xC0000001u;                            // count=1 | 32b idx | gather_mode
            g0.y = lds0 + (unsigned int)m * (TT * EE * 4); // lds_addr (bytes)
            g0.z = (unsigned int)base;                     // global_addr[31:0]
            g0.w = (unsigned int)((base >> 32) & 0x1FFFFFFu) | (2u << 30); // [56:32]|type=2
            v8i g1;
            g1[0] = 2 << 16;                               // data_size = 4B
            g1[1] = (int)((EE & 0xFFFF) << 16);            // tensor_dim0.lo16
            g1[2] = (int)((nwords & 0xFFFFu) << 16);       // dim0.hi=0 | tensor_dim1.lo16
            g1[3] = (int)(((nwords >> 16) & 0xFFFFu) | (EE << 16)); // dim1.hi | tile_dim0
            g1[4] = TT;                                    // tile_dim1 = #indices = 8
            g1[5] = EE;                                    // tensor_dim0_stride = 128 elems
            g1[6] = 0;
            g1[7] = 0;
            v4i g2, g3;
            g2[0] = ctx_bm[m * TT + 0]; g2[1] = ctx_bm[m * TT + 1];
            g2[2] = ctx_bm[m * TT + 2]; g2[3] = ctx_bm[m * TT + 3];
            g3[0] = ctx_bm[m * TT + 4]; g3[1] = ctx_bm[m * TT + 5];
            g3[2] = ctx_bm[m * TT + 6]; g3[3] = ctx_bm[m * TT + 7];
            TDM_LOAD(g0, g1, g2, g3);
        }
        __builtin_amdgcn_s_wait_tensorcnt(0);
    }
    __syncthreads();                       // staged rows visible to all waves
    for (int m = 0; m < MTILE; ++m) {      // branchless masked sum over T
        float acc = 0.0f;
#pragma unroll
        for (int t = 0; t < TT; ++t) {
            float v = rows[m][t][tid];
            acc += (ctx_bm[m * TT + t] != 0) ? v : 0.0f;
        }
        s_lds[m][tid] = acc;
    }
}

// Phase 1: p[b, m0..m0+15] = <s[b,m,:], q[b,:]>  via V_WMMA_F32_16X16X4_F32.
// A[i,k] = s[m0+i, ek+k], B[k,n] = q[ek+k] (replicated over n); extract col 0.
// NOTE: hot LDS (s_lds, q_lds) declared first => low offsets => DS immediates.
__global__ __launch_bounds__(128) void score_kernel(const int* __restrict__ ctx,
                                                    const float* __restrict__ Ch,
                                                    const float* __restrict__ q,
                                                    float* __restrict__ p,
                                                    unsigned int nwords) {
    __shared__ float s_lds[MTILE][EE + 4];  // hot: feeds WMMA via ds_load_2addr
    __shared__ float q_lds[EE];
    __shared__ float rows[MTILE][TT][EE];   // 64 KB TDM staging (cold, last)
    const int b   = blockIdx.x;
    const int m0  = blockIdx.y * MTILE;
    const int tid = threadIdx.x;

    q_lds[tid] = q[b * EE + tid];
    gather_tile_tdm(ctx + ((size_t)b * MM + m0) * TT, Ch, tid, nwords, rows, s_lds);
    __syncthreads();

    if (tid < 32) {                         // wave 0, EXEC all-ones inside
        const int lane = tid;
        const int half = lane >> 4;
        const int l16  = lane & 15;
        vf8 acc = {};
        for (int ek = 0; ek < EE; ek += 4) {
            vf2 a, bv;
            a.x  = s_lds[l16][ek + 2 * half + 0];
            a.y  = s_lds[l16][ek + 2 * half + 1];
            bv.x = q_lds[ek + 2 * half + 0];
            bv.y = q_lds[ek + 2 * half + 1];
            acc  = wmma_f32_16x16x4(a, bv, acc);
        }
        if (l16 == 0) {                     // D[i,0]: lane0 rows 0..7, lane16 rows 8..15
#pragma unroll
            for (int j = 0; j < 8; ++j)
                p[(size_t)b * MM + m0 + half * 8 + j] = acc[j];
        }
    }
}

// Phase 2: in-place softmax over M=512 per batch row.
__global__ __launch_bounds__(256) void softmax_kernel(float* __restrict__ p) {
    __shared__ float red[256];
    const int b = blockIdx.x, tid = threadIdx.x;
    float v0 = p[(size_t)b * MM + tid];
    float v1 = p[(size_t)b * MM + 256 + tid];

    red[tid] = fmaxf(v0, v1);
    __syncthreads();
    for (int s = 128; s > 0; s >>= 1) {
        if (tid < s) red[tid] = fmaxf(red[tid], red[tid + s]);
        __syncthreads();
    }
    const float mx = red[0];
    __syncthreads();

    float e0 = __expf(v0 - mx), e1 = __expf(v1 - mx);
    red[tid] = e0 + e1;
    __syncthreads();
    for (int s = 128; s > 0; s >>= 1) {
        if (tid < s) red[tid] += red[tid + s];
        __syncthreads();
    }
    const float inv = 1.0f / red[0];
    p[(size_t)b * MM + tid]       = e0 * inv;
    p[(size_t)b * MM + 256 + tid] = e1 * inv;
}

// Phase 3: partial o2 over a chunk of M.  A[i,k] = attn[m0+k] (replicated),
// B[k,n] = s2[m0+k, e0+n]; extract D row 0.  4 waves cover 128 e's.
__global__ __launch_bounds__(128) void weighted_sum_kernel(const int* __restrict__ ctx,
                                                           const float* __restrict__ Ch1,
                                                           const float* __restrict__ attn,
                                                           float* __restrict__ partial,
                                                           unsigned int nwords) {
    __shared__ float s_lds[MTILE][EE + 4];  // hot: feeds WMMA
    __shared__ float attn_lds[MTILE];
    __shared__ float rows[MTILE][TT][EE];   // 64 KB TDM staging (cold, last)
    const int b     = blockIdx.x;
    const int chunk = blockIdx.y;
    const int tid   = threadIdx.x;
    const int wave  = tid >> 5;
    const int lane  = tid & 31;
    const int half  = lane >> 4;
    const int l16   = lane & 15;
    const int e0    = wave * 32;

    vf8 acc0 = {}, acc1 = {};
    const int mbeg = chunk * (MM / MSPLIT);

    for (int m0 = mbeg; m0 < mbeg + MM / MSPLIT; m0 += MTILE) {
        __syncthreads();                    // previous tile fully consumed
        if (tid < MTILE) attn_lds[tid] = attn[(size_t)b * MM + m0 + tid];
        if (tid == 0 && m0 + MTILE < mbeg + MM / MSPLIT)
            __builtin_prefetch(ctx + ((size_t)b * MM + m0 + MTILE) * TT, 0, 0);
        gather_tile_tdm(ctx + ((size_t)b * MM + m0) * TT, Ch1, tid, nwords,
                        rows, s_lds);
        __syncthreads();

#pragma unroll
        for (int kc = 0; kc < MTILE; kc += 4) {
            vf2 a, b0, b1;
            a.x  = attn_lds[kc + 2 * half + 0];
            a.y  = attn_lds[kc + 2 * half + 1];
            b0.x = s_lds[kc + 2 * half + 0][e0 + l16];
            b0.y = s_lds[kc + 2 * half + 1][e0 + l16];
            b1.x = s_lds[kc + 2 * half + 0][e0 + 16 + l16];
            b1.y = s_lds[kc + 2 * half + 1][e0 + 16 + l16];
            acc0 = wmma_f32_16x16x4(a, b0, acc0);
            acc1 = wmma_f32_16x16x4(a, b1, acc1);
        }
    }
    if (lane < 16) {                        // D row 0: acc elem 0, lanes 0..15
        float* dst = partial + ((size_t)b * MSPLIT + chunk) * EE;
        dst[e0 + lane]      = acc0[0];
        dst[e0 + 16 + lane] = acc1[0];
    }
}

// Phase 4: sum MSPLIT partials, update q, emit o2 to d_out on last hop.
__global__ __launch_bounds__(128) void reduce_kernel(const float* __restrict__ partial,
                                                     float* __restrict__ q,
                                                     float* __restrict__ out,
                                                     int is_last) {
    const int b = blockIdx.x, tid = threadIdx.x;
    float s = 0.0f;
#pragma unroll
    for (int i = 0; i < MSPLIT; ++i)
        s += partial[((size_t)b * MSPLIT + i) * EE + tid];
    q[(size_t)b * EE + tid] += s;
    if (is_last) out[(size_t)b * EE + tid] = s;
}

extern "C" void kernel_launch(void* const* d_in, const int* in_sizes, int n_in,
                              void* d_out, int out_size, void* d_ws, size_t ws_size,
                              hipStream_t stream) {
    const int*   ctx = (const int*)d_in[0];               // [64,512,8] int32
    const float* C   = (const float*)d_in[1];             // [4, NWORDS, 128] f32
    const unsigned int nwords =
        (unsigned int)((size_t)in_sizes[1] / ((HOPSN + 1) * EE));

    float* q       = (float*)d_ws;                        // [64,128]
    float* p       = q + (size_t)BB * EE;                 // [64,512] scores/attn
    float* partial = p + (size_t)BB * MM;                 // [64,8,128]
    float* out     = (float*)d_out;                       // [64,128]

    hipMemsetAsync(q, 0, (size_t)BB * EE * sizeof(float), stream);

    for (int h = 0; h < HOPSN; ++h) {
        const float* Ch  = C + (size_t)h * nwords * EE;
        const float* Ch1 = C + (size_t)(h + 1) * nwords * EE;

        dim3 g1(BB, MM / MTILE);
        score_kernel<<<g1, 128, 0, stream>>>(ctx, Ch, q, p, nwords);
        softmax_kernel<<<BB, 256, 0, stream>>>(p);
        dim3 g3(BB, MSPLIT);
        weighted_sum_kernel<<<g3, 128, 0, stream>>>(ctx, Ch1, p, partial, nwords);
        reduce_kernel<<<BB, 128, 0, stream>>>(partial, q, out, h == HOPSN - 1);
    }
}